// SMDHeadInjection_29308856828003
// MI455X (gfx1250) — compile-verified
//
#include <hip/hip_runtime.h>
#include <math.h>

#define BATCH  8
#define SEQ    2048
#define DMODEL 1024
#define DK     64
#define NTOK   (BATCH*SEQ)   /* 16384 */
#define NSTEPS 4

typedef __attribute__((ext_vector_type(2))) float v2f;
typedef __attribute__((ext_vector_type(8))) float v8f;

#if defined(__HIP_DEVICE_COMPILE__) && !__has_builtin(__builtin_amdgcn_wmma_f32_16x16x4_f32)
#error "missing __builtin_amdgcn_wmma_f32_16x16x4_f32 on this toolchain"
#endif

static __device__ __forceinline__ v8f wmma_f32x4(v2f a, v2f b, v8f c) {
  // V_WMMA_F32_16X16X4_F32: D(16x16 f32) = A(16x4 f32) x B(4x16 f32) + C
  return __builtin_amdgcn_wmma_f32_16x16x4_f32(false, a, false, b, (short)0, c,
                                               false, false);
}

static __device__ __forceinline__ float sigm(float x) {
  return 1.0f / (1.0f + expf(-x));
}
static __device__ __forceinline__ float gelu_exact(float x) {
  return 0.5f * x * (1.0f + erff(x * 0.70710678118654752f));
}

// ---------------------------------------------------------------------------
// Kernel 1: per 16-token tile: h = gelu(X@W1^T + b1)  (WMMA, 8 col tiles)
//           V = X@wv^T                                 (WMMA, 4 col tiles)
//           features = sigmoid(h@w2^T + b2), Q, K, charge0 (scalar epilogue)
// ---------------------------------------------------------------------------
__global__ __launch_bounds__(256) void k_feat(
    const float* __restrict__ X,  const float* __restrict__ w1,
    const float* __restrict__ b1, const float* __restrict__ w2,
    const float* __restrict__ b2, const float* __restrict__ wq,
    const float* __restrict__ wk, const float* __restrict__ wc,
    const float* __restrict__ bc, const float* __restrict__ wv,
    float* __restrict__ Qb, float* __restrict__ Kb, float* __restrict__ Vb,
    float* __restrict__ c0)
{
  __shared__ float sX[16][68];    // stride 68 -> conflict-free A-frag reads
  __shared__ float sH[16][132];
  __shared__ float sF[16][28];
  const int tid = threadIdx.x;
  const int wid = tid >> 5, lane = tid & 31;
  const int hi = lane >> 4, lm = lane & 15;
  const int rowbase = blockIdx.x * 16;

  v8f acc0 = {0,0,0,0,0,0,0,0};   // h tile (cols wid*16..+15 of W1)
  v8f acc1 = {0,0,0,0,0,0,0,0};   // V tile (waves 0..3)
  const float* wB0 = w1 + (size_t)(wid*16 + lm) * DMODEL + hi*2;
  const float* wB1 = wv + (size_t)((wid & 3)*16 + lm) * DMODEL + hi*2;

  for (int kc = 0; kc < DMODEL/64; ++kc) {
    { // stage 16x64 X chunk: 4 floats per thread
      int r  = tid >> 4;
      int k4 = (tid & 15) * 4;
      const float4 v = *(const float4*)(X + (size_t)(rowbase + r)*DMODEL + kc*64 + k4);
      sX[r][k4+0] = v.x; sX[r][k4+1] = v.y; sX[r][k4+2] = v.z; sX[r][k4+3] = v.w;
    }
    __syncthreads();

    // batch all fragment loads for this 64-K chunk, then the WMMA chain:
    // keeps 16 global_load_b64 + 16 ds_load_b64 in flight (one wait, not 16)
    v2f af[16], b0f[16];
    #pragma unroll
    for (int k4 = 0; k4 < 16; ++k4) {
      const int kb = k4*4 + hi*2;
      af[k4].x = sX[lm][kb]; af[k4].y = sX[lm][kb+1];
      b0f[k4] = *(const v2f*)(wB0 + kc*64 + k4*4);
    }
    if (wid < 4) {           // uniform per-wave: specialized straight-line loop
      v2f b1f[16];
      #pragma unroll
      for (int k4 = 0; k4 < 16; ++k4)
        b1f[k4] = *(const v2f*)(wB1 + kc*64 + k4*4);
      #pragma unroll
      for (int k4 = 0; k4 < 16; ++k4) {
        acc0 = wmma_f32x4(af[k4], b0f[k4], acc0);
        acc1 = wmma_f32x4(af[k4], b1f[k4], acc1);
      }
    } else {
      #pragma unroll
      for (int k4 = 0; k4 < 16; ++k4)
        acc0 = wmma_f32x4(af[k4], b0f[k4], acc0);
    }
    __syncthreads();
  }

  { // h epilogue: bias + exact gelu -> LDS
    const int col = wid*16 + lm;
    const float bias = b1[col];
    #pragma unroll
    for (int r = 0; r < 8; ++r) {
      int m = r + hi*8;
      sH[m][col] = gelu_exact(acc0[r] + bias);
    }
  }
  if (wid < 4) { // V epilogue -> global
    const int col = wid*16 + lm;
    #pragma unroll
    for (int r = 0; r < 8; ++r) {
      int m = r + hi*8;
      Vb[(size_t)(rowbase + m)*DK + col] = acc1[r];
    }
  }
  __syncthreads();

  // features = sigmoid(h @ w2^T + b2)   (16x28, K=128)
  for (int idx = tid; idx < 16*28; idx += 256) {
    int m = idx / 28, c = idx % 28;
    const float* w2r = w2 + c*128;
    float s = b2[c];
    for (int k = 0; k < 128; ++k) s += sH[m][k] * w2r[k];
    sF[m][c] = sigm(s);
  }
  __syncthreads();

  // Q, K  (16x64, K=28)
  for (int idx = tid; idx < 16*DK; idx += 256) {
    int m = idx / DK, d = idx % DK;
    const float* wr = wq + d*28;
    float s = 0.f;
    #pragma unroll
    for (int c = 0; c < 28; ++c) s += sF[m][c] * wr[c];
    Qb[(size_t)(rowbase + m)*DK + d] = s;
  }
  for (int idx = tid; idx < 16*DK; idx += 256) {
    int m = idx / DK, d = idx % DK;
    const float* wr = wk + d*28;
    float s = 0.f;
    #pragma unroll
    for (int c = 0; c < 28; ++c) s += sF[m][c] * wr[c];
    Kb[(size_t)(rowbase + m)*DK + d] = s;
  }
  if (tid < 16) {
    float s = bc[0];
    #pragma unroll
    for (int c = 0; c < 28; ++c) s += sF[tid][c] * wc[c];
    c0[rowbase + tid] = sigm(s);
  }
}

// ---------------------------------------------------------------------------
// Kernel 2: compat = Q K^T / 8, causal tiles only (one wave per 16x16 tile)
// ---------------------------------------------------------------------------
__global__ __launch_bounds__(256) void k_compat(
    const float* __restrict__ Qb, const float* __restrict__ Kb,
    float* __restrict__ compat)
{
  const int tid = threadIdx.x;
  const int wid = tid >> 5, lane = tid & 31;
  const int hi = lane >> 4, lm = lane & 15;
  const int b  = blockIdx.x >> 7;
  const int it = blockIdx.x & 127;
  const int jt = blockIdx.y * 8 + wid;
  if (jt > it) return;
  const float* qrow = Qb + (size_t)(b*SEQ + it*16 + lm) * DK + hi*2;
  const float* krow = Kb + (size_t)(b*SEQ + jt*16 + lm) * DK + hi*2;
  v2f aq[16], bk[16];
  #pragma unroll
  for (int k4 = 0; k4 < 16; ++k4) {     // batch 32 b64 loads, single wait
    aq[k4] = *(const v2f*)(qrow + k4*4);
    bk[k4] = *(const v2f*)(krow + k4*4);
  }
  v8f acc = {0,0,0,0,0,0,0,0};
  #pragma unroll
  for (int k4 = 0; k4 < 16; ++k4)
    acc = wmma_f32x4(aq[k4], bk[k4], acc);
  float* crow = compat + (size_t)b * SEQ * SEQ;
  #pragma unroll
  for (int r = 0; r < 8; ++r) {
    int m = r + hi*8;
    crow[(size_t)(it*16 + m)*SEQ + jt*16 + lm] = acc[r] * 0.125f;
  }
}

__global__ void k_zero(float* __restrict__ p, int n) {
  int i = blockIdx.x * blockDim.x + threadIdx.x;
  if (i < n) p[i] = 0.f;
}

// ---------------------------------------------------------------------------
// Kernel 3: iteration t: row softmax of compat*(1+step*Sum c_tau c_tau^T),
//           column-sum into received[]. One wave per row; LDS-staged atomics.
// ---------------------------------------------------------------------------
__global__ __launch_bounds__(256) void k_iter(
    const float* __restrict__ compat, const float* __restrict__ cb,
    float* __restrict__ received, const float* __restrict__ step_p, int t)
{
  __shared__ float srecv[SEQ];
  const int tid = threadIdx.x;
  const int wid = tid >> 5, lane = tid & 31;
  const int row = blockIdx.x * 8 + wid;   // global token; block stays in one batch
  const int b = row >> 11, i = row & 2047;
  const float step = step_p[0];
  for (int j = tid; j < SEQ; j += 256) srecv[j] = 0.f;
  __syncthreads();

  const float* crow = compat + (size_t)row * SEQ;
  const float* cjs[3]; float cis[3];
  for (int tau = 1; tau <= t; ++tau) {
    cjs[tau-1] = cb + (size_t)tau*NTOK + b*SEQ;
    cis[tau-1] = cjs[tau-1][i];
  }
  // pass 1: online max / sumexp
  float m = -3.0e38f, s = 0.f;
  for (int j = lane; j <= i; j += 32) {
    float f = 1.f;
    for (int tau = 0; tau < t; ++tau) f += step * cis[tau] * cjs[tau][j];
    float x = crow[j] * f;
    float mn = fmaxf(m, x);
    s = s * expf(m - mn) + expf(x - mn);
    m = mn;
  }
  #pragma unroll
  for (int off = 16; off > 0; off >>= 1) {
    float m2 = __shfl_xor(m, off);
    float s2 = __shfl_xor(s, off);
    float mn = fmaxf(m, m2);
    s = s * expf(m - mn) + s2 * expf(m2 - mn);
    m = mn;
  }
  const float inv = 1.0f / s;
  // pass 2: attn values -> LDS column accumulators
  for (int j = lane; j <= i; j += 32) {
    float f = 1.f;
    for (int tau = 0; tau < t; ++tau) f += step * cis[tau] * cjs[tau][j];
    float x = crow[j] * f;
    atomicAdd(&srecv[j], expf(x - m) * inv);
  }
  __syncthreads();
  const int imax = (blockIdx.x * 8 + 7) & 2047;
  for (int j = tid; j <= imax; j += 256)
    atomicAdd(&received[b*SEQ + j], srecv[j]);
}

__global__ void k_charge(const float* __restrict__ cin, float* __restrict__ cout,
                         const float* __restrict__ received,
                         const float* __restrict__ decay_p)
{
  int i = blockIdx.x * blockDim.x + threadIdx.x;
  if (i >= NTOK) return;
  float sg = sigm(received[i] - 1.0f);
  cout[i] = cin[i] * (1.0f - decay_p[0] * sg);
}

// ---------------------------------------------------------------------------
// Kernel 4: final softmax fused with attn@V (flash-style, WMMA P@V)
// ---------------------------------------------------------------------------
__global__ __launch_bounds__(128) void k_attnv(
    const float* __restrict__ compat, const float* __restrict__ cb,
    const float* __restrict__ Vb, float* __restrict__ O1,
    const float* __restrict__ step_p)
{
  __shared__ float sm[16], sinv[16];
  __shared__ float sP[16][20];
  const int tid = threadIdx.x;
  const int wid = tid >> 5, lane = tid & 31;
  const int hi = lane >> 4, lm = lane & 15;
  const int b = blockIdx.x >> 7, it = blockIdx.x & 127;
  const float step = step_p[0];
  const float* c1 = cb + 1*(size_t)NTOK + b*SEQ;
  const float* c2 = cb + 2*(size_t)NTOK + b*SEQ;
  const float* c3 = cb + 3*(size_t)NTOK + b*SEQ;
  const float* c4 = cb + 4*(size_t)NTOK + b*SEQ;

  // stage A: per-row max & 1/Z (wave w -> rows w*4..w*4+3)
  for (int rr = 0; rr < 4; ++rr) {
    const int m = wid*4 + rr;
    const int i = it*16 + m;
    const float* crow = compat + (size_t)(b*SEQ + i) * SEQ;
    const float ci1 = c1[i], ci2 = c2[i], ci3 = c3[i], ci4 = c4[i];
    float mx = -3.0e38f, s = 0.f;
    for (int j = lane; j <= i; j += 32) {
      float f = 1.f + step*(ci1*c1[j] + ci2*c2[j] + ci3*c3[j] + ci4*c4[j]);
      float x = crow[j] * f;
      float mn = fmaxf(mx, x);
      s = s * expf(mx - mn) + expf(x - mn);
      mx = mn;
    }
    #pragma unroll
    for (int off = 16; off > 0; off >>= 1) {
      float m2 = __shfl_xor(mx, off);
      float s2 = __shfl_xor(s, off);
      float mn = fmaxf(mx, m2);
      s = s * expf(mx - mn) + s2 * expf(m2 - mn);
      mx = mn;
    }
    if (lane == 0) { sm[m] = mx; sinv[m] = 1.0f / s; }
  }
  __syncthreads();

  v8f acc = {0,0,0,0,0,0,0,0};
  const int n0 = wid * 16;                    // V-dim tile for this wave
  for (int jt = 0; jt <= it; ++jt) {
    for (int e = tid; e < 256; e += 128) {    // P tile 16x16 -> LDS
      int m = e >> 4, j = e & 15;
      int ig = it*16 + m, jg = jt*16 + j;
      float p = 0.f;
      if (jg <= ig) {
        float f = 1.f + step*(c1[ig]*c1[jg] + c2[ig]*c2[jg]
                            + c3[ig]*c3[jg] + c4[ig]*c4[jg]);
        float x = compat[(size_t)(b*SEQ + ig)*SEQ + jg] * f;
        p = expf(x - sm[m]) * sinv[m];
      }
      sP[m][j] = p;
    }
    __syncthreads();
    const float* vbase = Vb + (size_t)(b*SEQ + jt*16)*DK + n0 + lm;
    v2f af[4], bf[4];
    #pragma unroll
    for (int kk = 0; kk < 4; ++kk) {          // batch loads, then wmma chain
      int kb = kk*4 + hi*2;
      af[kk].x = sP[lm][kb];               af[kk].y = sP[lm][kb+1];
      bf[kk].x = vbase[(size_t)kb*DK];     bf[kk].y = vbase[(size_t)(kb+1)*DK];
    }
    #pragma unroll
    for (int kk = 0; kk < 4; ++kk)
      acc = wmma_f32x4(af[kk], bf[kk], acc);
    __syncthreads();
  }
  #pragma unroll
  for (int r = 0; r < 8; ++r) {
    int m = r + hi*8;
    O1[(size_t)(b*SEQ + it*16 + m)*DK + n0 + lm] = acc[r];
  }
}

// ---------------------------------------------------------------------------
// Kernel 5: out = alpha * (O1 @ wo^T)   (WMMA, K=64)
// ---------------------------------------------------------------------------
__global__ __launch_bounds__(256) void k_out(
    const float* __restrict__ O1, const float* __restrict__ wo,
    const float* __restrict__ mix_p, float* __restrict__ out)
{
  const int tid = threadIdx.x;
  const int wid = tid >> 5, lane = tid & 31;
  const int hi = lane >> 4, lm = lane & 15;
  const int rt = blockIdx.x;
  const int nt = blockIdx.y * 8 + wid;
  const float alpha = sigm(mix_p[0]) * 0.2f;
  const float* arow = O1 + (size_t)(rt*16 + lm)*DK + hi*2;
  const float* brow = wo + (size_t)(nt*16 + lm)*DK + hi*2;
  v2f af[16], bf[16];
  #pragma unroll
  for (int k4 = 0; k4 < 16; ++k4) {
    af[k4] = *(const v2f*)(arow + k4*4);
    bf[k4] = *(const v2f*)(brow + k4*4);
  }
  v8f acc = {0,0,0,0,0,0,0,0};
  #pragma unroll
  for (int k4 = 0; k4 < 16; ++k4)
    acc = wmma_f32x4(af[k4], bf[k4], acc);
  #pragma unroll
  for (int r = 0; r < 8; ++r) {
    int m = r + hi*8;
    out[(size_t)(rt*16 + m)*DMODEL + nt*16 + lm] = acc[r] * alpha;
  }
}

// ---------------------------------------------------------------------------
extern "C" void kernel_launch(void* const* d_in, const int* in_sizes, int n_in,
                              void* d_out, int out_size, void* d_ws, size_t ws_size,
                              hipStream_t stream)
{
  (void)in_sizes; (void)n_in; (void)out_size; (void)ws_size;
  const float* hs      = (const float*)d_in[0];
  // d_in[1] = attention_mask (all ones; unused by the reference math)
  const float* w1      = (const float*)d_in[2];
  const float* b1      = (const float*)d_in[3];
  const float* w2      = (const float*)d_in[4];
  const float* b2      = (const float*)d_in[5];
  const float* wq      = (const float*)d_in[6];
  const float* wk      = (const float*)d_in[7];
  const float* wc      = (const float*)d_in[8];
  const float* bc      = (const float*)d_in[9];
  const float* step_p  = (const float*)d_in[10];
  const float* decay_p = (const float*)d_in[11];
  const float* wv      = (const float*)d_in[12];
  const float* wo      = (const float*)d_in[13];
  const float* mix_p   = (const float*)d_in[14];
  float* out = (float*)d_out;

  char* ws = (char*)d_ws;
  size_t off = 0;
  auto alloc = [&](size_t bytes) -> void* {
    void* p = ws + off; off += (bytes + 255) & ~(size_t)255; return p;
  };
  float* Qb     = (float*)alloc((size_t)NTOK * DK * 4);
  float* Kb     = (float*)alloc((size_t)NTOK * DK * 4);
  float* Vb     = (float*)alloc((size_t)NTOK * DK * 4);
  float* O1     = (float*)alloc((size_t)NTOK * DK * 4);
  float* cb     = (float*)alloc((size_t)5 * NTOK * 4);   // c0..c4
  float* recv   = (float*)alloc((size_t)NTOK * 4);
  float* compat = (float*)alloc((size_t)NTOK * SEQ * 4); // 134 MB, L2-resident

  k_feat<<<NTOK/16, 256, 0, stream>>>(hs, w1, b1, w2, b2, wq, wk, wc, bc, wv,
                                      Qb, Kb, Vb, cb /* c0 */);
  k_compat<<<dim3(BATCH*128, 16), 256, 0, stream>>>(Qb, Kb, compat);
  for (int t = 0; t < NSTEPS; ++t) {
    k_zero<<<NTOK/256, 256, 0, stream>>>(recv, NTOK);
    k_iter<<<NTOK/8, 256, 0, stream>>>(compat, cb, recv, step_p, t);
    k_charge<<<NTOK/256, 256, 0, stream>>>(cb + (size_t)t*NTOK,
                                           cb + (size_t)(t+1)*NTOK,
                                           recv, decay_p);
  }
  k_attnv<<<BATCH*128, 128, 0, stream>>>(compat, cb, Vb, O1, step_p);
  k_out<<<dim3(NTOK/16, 8), 256, 0, stream>>>(O1, wo, mix_p, out);
}